// InferModel_73083163509446
// MI455X (gfx1250) — compile-verified
//
#include <hip/hip_runtime.h>
#include <cstdint>

#define TPB1 1024
#define TILE 4096      // floats per async tile = TPB1 * 4
#define TPB2 256
#define CCAP 2048      // candidate cap per row
#define MCAP 1024      // survivor cap per row
#define BINS 4096      // 12-bit ordered-key histogram

__device__ __forceinline__ unsigned rotl32(unsigned x, unsigned r) {
  return (x << r) | (x >> (32u - r));
}

// order-preserving float -> uint key (ascending)
__device__ __forceinline__ unsigned ordered_key(float f) {
  unsigned u = __float_as_uint(f);
  return u ^ ((u >> 31) ? 0xFFFFFFFFu : 0x80000000u);
}

// Low 32 bits of a generic pointer into LDS == LDS byte offset (shared aperture layout)
__device__ __forceinline__ unsigned lds_off(const void* p) {
  return (unsigned)(uintptr_t)p;
}

// ---- JAX threefry2x32-20, key = PRNGKey(42) = (0, 42); gumbel noise at flat index ----
__device__ __forceinline__ float gumbel_at(long long flat, long long half) {
  const unsigned k0 = 0u, k1 = 42u;
  unsigned c0, c1; bool hi;
  if (flat < half) { c0 = (unsigned)flat;          c1 = (unsigned)(flat + half); hi = false; }
  else             { c0 = (unsigned)(flat - half); c1 = (unsigned)flat;          hi = true;  }
  unsigned ks2 = 0x1BD11BDAu ^ k0 ^ k1;
  unsigned x0 = c0 + k0, x1 = c1 + k1;
#define TF_R(r) { x0 += x1; x1 = rotl32(x1, r); x1 ^= x0; }
  TF_R(13) TF_R(15) TF_R(26) TF_R(6)   x0 += k1;  x1 += ks2 + 1u;
  TF_R(17) TF_R(29) TF_R(16) TF_R(24)  x0 += ks2; x1 += k0 + 2u;
  TF_R(13) TF_R(15) TF_R(26) TF_R(6)   x0 += k0;  x1 += k1 + 3u;
  TF_R(17) TF_R(29) TF_R(16) TF_R(24)  x0 += k1;  x1 += ks2 + 4u;
  TF_R(13) TF_R(15) TF_R(26) TF_R(6)   x0 += ks2; x1 += k0 + 5u;
#undef TF_R
  unsigned bits = hi ? x1 : x0;
  float f = __uint_as_float((bits >> 9) | 0x3F800000u) - 1.0f;  // [0,1)
  f = fmaxf(f, 1.1754944e-38f);
  return -__logf(-__logf(f));
}

// ============ Kernel 1: per-row histogram select + zero-fill + compact ============
__global__ __launch_bounds__(TPB1) void topk_select_kernel(
    const float* __restrict__ logits, const float* __restrict__ temp,
    const int* __restrict__ topk_p, float* __restrict__ probs,
    float* __restrict__ candVal, int* __restrict__ candIdx,
    int* __restrict__ candCnt, int V)
{
  __shared__ unsigned hist[BINS];
  __shared__ float stage[2][TILE];
  __shared__ unsigned s1[TPB1];
  __shared__ unsigned sT;
  __shared__ unsigned scnt;

  const int row = blockIdx.x;
  const int tid = threadIdx.x;
  const float invT = 1.0f / temp[0];
  const float* rowp = logits + (size_t)row * (size_t)V;
  const int ntiles = V / TILE;
  const int tailBase = ntiles * TILE;
  const int tailN = V - tailBase;

  for (int i = tid; i < BINS; i += TPB1) hist[i] = 0u;
  if (tid == 0) scnt = 0u;
  __syncthreads();

  // ---- Phase A: async-to-LDS double-buffered streaming histogram ----
  const unsigned ldsA = lds_off(&stage[0][tid * 4]);
  const unsigned ldsB = lds_off(&stage[1][tid * 4]);
  if (ntiles > 0) {
    unsigned off = (unsigned)(tid * 4) * 4u;
    asm volatile("global_load_async_to_lds_b128 %0, %1, %2"
                 :: "v"(ldsA), "v"(off), "s"(rowp) : "memory");
  }
  for (int t = 0; t < ntiles; ++t) {
    if (t + 1 < ntiles) {
      unsigned off = (unsigned)((t + 1) * TILE + tid * 4) * 4u;
      unsigned dst = (t & 1) ? ldsA : ldsB;
      asm volatile("global_load_async_to_lds_b128 %0, %1, %2"
                   :: "v"(dst), "v"(off), "s"(rowp) : "memory");
      asm volatile("s_wait_asynccnt 1" ::: "memory");   // tile t landed
    } else {
      asm volatile("s_wait_asynccnt 0" ::: "memory");
    }
    const float* sp = &stage[t & 1][tid * 4];
#pragma unroll
    for (int j = 0; j < 4; ++j) {
      unsigned key = ordered_key(sp[j] * invT);
      atomicAdd(&hist[key >> 20], 1u);
    }
  }
  if (tid * 4 < tailN) {   // remainder without staging
    float4 x = *(const float4*)(rowp + tailBase + tid * 4);
    atomicAdd(&hist[ordered_key(x.x * invT) >> 20], 1u);
    atomicAdd(&hist[ordered_key(x.y * invT) >> 20], 1u);
    atomicAdd(&hist[ordered_key(x.z * invT) >> 20], 1u);
    atomicAdd(&hist[ordered_key(x.w * invT) >> 20], 1u);
  }
  __syncthreads();

  // ---- Find bin containing the K-th largest (block scan over 4096 bins) ----
  unsigned v0 = hist[4*tid+0], v1 = hist[4*tid+1], v2 = hist[4*tid+2], v3 = hist[4*tid+3];
  unsigned mySum = v0 + v1 + v2 + v3;
  s1[tid] = mySum;
  __syncthreads();
  for (int d = 1; d < TPB1; d <<= 1) {
    unsigned add = (tid >= d) ? s1[tid - d] : 0u;
    __syncthreads();
    s1[tid] += add;
    __syncthreads();
  }
  const unsigned K = (unsigned)topk_p[0];
  unsigned total = s1[TPB1 - 1];
  unsigned suffIncl = total - s1[tid] + mySum;   // count in bins >= 4*tid
  unsigned above = suffIncl - mySum;             // count in bins >  4*tid+3
  unsigned c3 = above + v3, c2 = c3 + v2, c1 = c2 + v1, c0 = c1 + v0;
  if      (above < K && c3 >= K) sT = 4*tid + 3;
  else if (c3 < K && c2 >= K)    sT = 4*tid + 2;
  else if (c2 < K && c1 >= K)    sT = 4*tid + 1;
  else if (c1 < K && c0 >= K)    sT = 4*tid + 0;
  __syncthreads();
  const unsigned Tbin = sT;

  // ---- Phase B: zero probs row (bulk of output), compact candidates (L2 re-read) ----
  float* prow = probs + (size_t)row * (size_t)V;
  const float4 z4 = make_float4(0.f, 0.f, 0.f, 0.f);
  for (int base = 0; base < V; base += TPB1 * 4) {
    int i = base + tid * 4;
    if (i < V) {
      float4 x = *(const float4*)(rowp + i);
      *(float4*)(prow + i) = z4;
      float vals[4] = {x.x, x.y, x.z, x.w};
#pragma unroll
      for (int j = 0; j < 4; ++j) {
        float s = vals[j] * invT;
        if ((ordered_key(s) >> 20) >= Tbin) {
          unsigned slot = atomicAdd(&scnt, 1u);
          if (slot < CCAP) {
            candVal[row * CCAP + slot] = s;
            candIdx[row * CCAP + slot] = i + j;
          }
        }
      }
    }
  }
  __syncthreads();
  if (tid == 0) candCnt[row] = (int)(scnt < CCAP ? scnt : CCAP);
}

// ============ Kernel 2: exact kth, top-p, softmax scatter, gumbel-argmax sample ============
__global__ __launch_bounds__(TPB2) void sample_kernel(
    const float* __restrict__ candVal, const int* __restrict__ candIdx,
    const int* __restrict__ candCnt, const float* __restrict__ topp_p,
    const int* __restrict__ topk_p, float* __restrict__ probs,
    int* __restrict__ token, int V, long long half)
{
  __shared__ float cv[CCAP];
  __shared__ int   ci[CCAP];
  __shared__ float sv[MCAP];
  __shared__ int   si[MCAP];
  __shared__ float se[MCAP];
  __shared__ unsigned char kf[MCAP];
  __shared__ float redf[TPB2];
  __shared__ unsigned redi[TPB2];
  __shared__ float kthVal, smax, stot, skept;
  __shared__ unsigned mcnt;

  const int row = blockIdx.x;
  const int tid = threadIdx.x;
  const int n = min(candCnt[row], CCAP);
  const int K = topk_p[0];
  const float topp = topp_p[0];

  if (n == 0) { if (tid == 0) token[row] = 0; return; }

  for (int i = tid; i < n; i += TPB2) {
    cv[i] = candVal[row * CCAP + i];
    ci[i] = candIdx[row * CCAP + i];
  }
  if (tid == 0) { mcnt = 0u; kthVal = -INFINITY; }
  __syncthreads();

  // exact K-th largest value among candidates (rank counting; ties kept like lax.top_k)
  if (n > K) {
    for (int i = tid; i < n; i += TPB2) {
      float v = cv[i];
      int gt = 0, ge = 0;
      for (int j = 0; j < n; ++j) {
        float w = cv[j];
        gt += (w > v);
        ge += (w >= v);
      }
      if (gt < K && ge >= K) kthVal = v;   // unique value; benign same-value race
    }
  }
  __syncthreads();
  const float kth = kthVal;

  for (int i = tid; i < n; i += TPB2) {
    if (cv[i] >= kth) {
      unsigned s = atomicAdd(&mcnt, 1u);
      if (s < MCAP) { sv[s] = cv[i]; si[s] = ci[i]; }
    }
  }
  __syncthreads();
  const int m = (int)(mcnt < MCAP ? mcnt : MCAP);

  // max over survivors
  float lm = -INFINITY;
  for (int i = tid; i < m; i += TPB2) lm = fmaxf(lm, sv[i]);
  redf[tid] = lm; __syncthreads();
  for (int s = TPB2 / 2; s > 0; s >>= 1) {
    if (tid < s) redf[tid] = fmaxf(redf[tid], redf[tid + s]);
    __syncthreads();
  }
  if (tid == 0) smax = redf[0];
  __syncthreads();
  const float mx = smax;

  // exp + total mass (softmax over top-k filtered scores)
  float ls = 0.f;
  for (int i = tid; i < m; i += TPB2) { float e = __expf(sv[i] - mx); se[i] = e; ls += e; }
  redf[tid] = ls; __syncthreads();
  for (int s = TPB2 / 2; s > 0; s >>= 1) {
    if (tid < s) redf[tid] += redf[tid + s];
    __syncthreads();
  }
  if (tid == 0) stot = redf[0];
  __syncthreads();
  const float tot = stot;

  // stable ascending cumulative; remove tail with cumprob <= 1-p; kept mass
  float lk = 0.f;
  for (int i = tid; i < m; i += TPB2) {
    float v = sv[i]; int ii = si[i];
    float c = 0.f;
    for (int j = 0; j < m; ++j) {
      float w = sv[j];
      if (w < v || (w == v && si[j] <= ii)) c += se[j];
    }
    bool keep = (c / tot) > (1.0f - topp);
    kf[i] = keep ? 1 : 0;
    if (keep) lk += se[i];
  }
  redf[tid] = lk; __syncthreads();
  for (int s = TPB2 / 2; s > 0; s >>= 1) {
    if (tid < s) redf[tid] += redf[tid + s];
    __syncthreads();
  }
  if (tid == 0) skept = redf[0];
  __syncthreads();
  const float sk = skept;

  // scatter final probs + threefry-gumbel argmax over kept set
  float bestv = -INFINITY; unsigned besti = 0x7FFFFFFFu;
  for (int i = tid; i < m; i += TPB2) {
    if (kf[i]) {
      probs[(size_t)row * (size_t)V + (size_t)si[i]] = se[i] / sk;
      long long flat = (long long)row * (long long)V + (long long)si[i];
      float z = sv[i] + gumbel_at(flat, half);
      unsigned ui = (unsigned)si[i];
      if (z > bestv || (z == bestv && ui < besti)) { bestv = z; besti = ui; }
    }
  }
  redf[tid] = bestv; redi[tid] = besti; __syncthreads();
  for (int s = TPB2 / 2; s > 0; s >>= 1) {
    if (tid < s) {
      float ov = redf[tid + s]; unsigned oi = redi[tid + s];
      if (ov > redf[tid] || (ov == redf[tid] && oi < redi[tid])) { redf[tid] = ov; redi[tid] = oi; }
    }
    __syncthreads();
  }
  if (tid == 0) token[row] = (int)redi[0];
}

extern "C" void kernel_launch(void* const* d_in, const int* in_sizes, int n_in,
                              void* d_out, int out_size, void* d_ws, size_t ws_size,
                              hipStream_t stream) {
  const float* logits = (const float*)d_in[0];
  const float* temp   = (const float*)d_in[1];
  const float* topp   = (const float*)d_in[2];
  const int*   topk   = (const int*)d_in[3];

  const long long NV = (long long)in_sizes[0];   // B*V
  const int B = out_size - (int)NV;              // out = B tokens ++ B*V probs
  const int V = (int)(NV / (long long)B);

  int*   token = (int*)d_out;
  float* probs = (float*)d_out + B;

  float* candVal = (float*)d_ws;                       // B*CCAP floats
  int*   candIdx = (int*)(candVal + (size_t)B * CCAP); // B*CCAP ints
  int*   candCnt = candIdx + (size_t)B * CCAP;         // B ints

  topk_select_kernel<<<B, TPB1, 0, stream>>>(logits, temp, topk, probs,
                                             candVal, candIdx, candCnt, V);
  sample_kernel<<<B, TPB2, 0, stream>>>(candVal, candIdx, candCnt, topp, topk,
                                        probs, token, V, NV / 2);
}